// GNNStack_15899968930260
// MI455X (gfx1250) — compile-verified
//
#include <hip/hip_runtime.h>

#define NN 50000
#define EE 800000
#define HH 64

typedef float v2f __attribute__((ext_vector_type(2)));
typedef float v8f __attribute__((ext_vector_type(8)));

// ---------------------------------------------------------------- utilities
__global__ __launch_bounds__(256) void fill_zero_kernel(float* __restrict__ p, int n) {
    int i = blockIdx.x * 256 + threadIdx.x;
    if (i < n) p[i] = 0.0f;
}

__global__ __launch_bounds__(256) void fill_zero4_kernel(float4* __restrict__ p, int n4) {
    int i = blockIdx.x * 256 + threadIdx.x;
    if (i < n4) p[i] = make_float4(0.f, 0.f, 0.f, 0.f);
}

__global__ __launch_bounds__(256) void degree_kernel(const int* __restrict__ dst,
                                                     float* __restrict__ deg, int ne) {
    int i = blockIdx.x * 256 + threadIdx.x;
    if (i < ne) atomicAdd(&deg[dst[i]], 1.0f);
}

// dis[i] = (deg+1)^-1/2 ; inv[i] = 1/(deg+1)
__global__ __launch_bounds__(256) void norm_kernel(float* __restrict__ dis,
                                                   float* __restrict__ inv, int n) {
    int i = blockIdx.x * 256 + threadIdx.x;
    if (i < n) {
        float d = dis[i] + 1.0f;
        dis[i] = rsqrtf(d);
        inv[i] = 1.0f / d;
    }
}

// ---------------------------------------------------------------- WMMA GEMM
// Out[nrows x 64] = A[nrows x 64] @ W[64 x 64] (+ bias broadcast over rows)
// 256 threads = 8 waves; each wave owns one 16x16 output tile.
// Boundary handling: A row index is CLAMPED (not masked) -- an A lane only
// feeds output row M, and rows >= nrows are never stored, so clamped-row
// garbage is harmless.  This keeps every load unconditional and EXEC all-1s
// through the whole WMMA pipeline (no exec-masked scalar loads).
__global__ __launch_bounds__(256) void gemm64_wmma_kernel(const float* __restrict__ A,
                                                          const float* __restrict__ W,
                                                          const float* __restrict__ bias,
                                                          float* __restrict__ Out,
                                                          int nrows) {
    __shared__ float wlds[HH * HH];   // 16 KB of 320 KB/WGP
    {
        const float4* w4 = (const float4*)W;
        float4* l4 = (float4*)wlds;
        for (int i = threadIdx.x; i < (HH * HH) / 4; i += 256) l4[i] = w4[i];
    }
    __syncthreads();

    const int lane   = threadIdx.x & 31;
    const int wave   = threadIdx.x >> 5;
    const int lane15 = lane & 15;
    const int laneHi = lane >> 4;                 // A/B: K half; C/D: M / M+8
    const int rowBase = blockIdx.x * 32 + (wave >> 2) * 16;
    const int colBase = (wave & 3) * 16;

    int aRow = rowBase + lane15;
    if (aRow > nrows - 1) aRow = nrows - 1;       // clamp, never mask
    const float* __restrict__ arow = A + (size_t)aRow * HH;

    // Preload all 16 A fragments (one load clause of global_load_b64).
    v2f afrag[16];
#pragma unroll
    for (int i = 0; i < 16; ++i)
        afrag[i] = *(const v2f*)(arow + 4 * i + laneHi * 2);

    v8f c = {};
#pragma unroll
    for (int i = 0; i < 16; ++i) {
        const int kk = 4 * i + laneHi * 2;
        v2f b;
        b.x = wlds[kk * HH + colBase + lane15];
        b.y = wlds[(kk + 1) * HH + colBase + lane15];
        // (neg_a, A, neg_b, B, c_mod, C, reuse_a, reuse_b)
        c = __builtin_amdgcn_wmma_f32_16x16x4_f32(false, afrag[i], false, b,
                                                  (short)0, c, false, false);
    }

#pragma unroll
    for (int v = 0; v < 8; ++v) {
        int row = rowBase + laneHi * 8 + v;
        if (row < nrows) {
            int col = colBase + lane15;
            float val = c[v] + (bias ? bias[col] : 0.0f);
            Out[(size_t)row * HH + col] = val;
        }
    }
}

// ------------------------------------------------------- edge scatter (agg)
// 16 threads per edge, each handles 4 consecutive features (float4 read,
// 4 float atomics into agg[dst]).  agg[d] += t[s] * dis[s] * dis[d]
__global__ __launch_bounds__(256) void scatter_kernel(const int* __restrict__ src,
                                                      const int* __restrict__ dst,
                                                      const float* __restrict__ dis,
                                                      const float* __restrict__ t,
                                                      float* __restrict__ agg, int ne) {
    int tid = blockIdx.x * 256 + threadIdx.x;
    int e = tid >> 4;
    if (e >= ne) return;
    int f4 = (tid & 15) * 4;
    int s = src[e], d = dst[e];
    float coef = dis[s] * dis[d];
    const float4 tv = *(const float4*)(t + (size_t)s * HH + f4);
    float* ap = agg + (size_t)d * HH + f4;
    atomicAdd(ap + 0, tv.x * coef);
    atomicAdd(ap + 1, tv.y * coef);
    atomicAdd(ap + 2, tv.z * coef);
    atomicAdd(ap + 3, tv.w * coef);
}

// -------------------------------------------- self-loop + bias + relu fuse
// One float4 per thread: val = agg + t*inv[node] + b[f]; emb?=val; h=relu(val)
__global__ __launch_bounds__(256) void finalize_kernel(const float4* __restrict__ t,
                                                       const float4* __restrict__ agg,
                                                       const float* __restrict__ inv,
                                                       const float4* __restrict__ bias4,
                                                       float4* __restrict__ h,
                                                       float4* __restrict__ emb, int n) {
    int tid = blockIdx.x * 256 + threadIdx.x;      // over n * 16 float4s
    if (tid >= n * (HH / 4)) return;
    int i = tid >> 4;                               // node
    int f = tid & 15;                               // float4 within row
    float iv = inv[i];
    float4 tv = t[tid], av = agg[tid], bv = bias4[f];
    float4 val;
    val.x = av.x + tv.x * iv + bv.x;
    val.y = av.y + tv.y * iv + bv.y;
    val.z = av.z + tv.z * iv + bv.z;
    val.w = av.w + tv.w * iv + bv.w;
    if (emb) emb[tid] = val;
    float4 r;
    r.x = fmaxf(val.x, 0.f); r.y = fmaxf(val.y, 0.f);
    r.z = fmaxf(val.z, 0.f); r.w = fmaxf(val.w, 0.f);
    h[tid] = r;
}

// ----------------------------------------------- 64->2 head + log_softmax
__global__ __launch_bounds__(256) void head_kernel(const float* __restrict__ y1,
                                                   const float* __restrict__ Wp2,
                                                   const float* __restrict__ bp2,
                                                   float* __restrict__ out, int n) {
    int i = blockIdx.x * 256 + threadIdx.x;
    if (i >= n) return;
    const float* yi = y1 + (size_t)i * HH;
    float a0 = bp2[0], a1 = bp2[1];
#pragma unroll 8
    for (int k = 0; k < HH; ++k) {
        float v = yi[k];
        a0 += v * Wp2[k * 2 + 0];
        a1 += v * Wp2[k * 2 + 1];
    }
    float m = fmaxf(a0, a1);
    float lse = m + logf(expf(a0 - m) + expf(a1 - m));
    out[i * 2 + 0] = a0 - lse;
    out[i * 2 + 1] = a1 - lse;
}

// ---------------------------------------------------------------- launcher
extern "C" void kernel_launch(void* const* d_in, const int* in_sizes, int n_in,
                              void* d_out, int out_size, void* d_ws, size_t ws_size,
                              hipStream_t stream) {
    const float* x   = (const float*)d_in[0];
    const int*   ei  = (const int*)d_in[1];   // [2, E] flat: src then dst
    const float* W1  = (const float*)d_in[2];
    const float* b1  = (const float*)d_in[3];
    const float* W2  = (const float*)d_in[4];
    const float* b2  = (const float*)d_in[5];
    const float* W3  = (const float*)d_in[6];
    const float* b3  = (const float*)d_in[7];
    const float* Wp1 = (const float*)d_in[8];
    const float* bp1 = (const float*)d_in[9];
    const float* Wp2 = (const float*)d_in[10];
    const float* bp2 = (const float*)d_in[11];

    const int* src = ei;
    const int* dst = ei + EE;

    // workspace layout (floats): dis[N] inv[N] h[N*64] t[N*64] agg[N*64]
    float* ws  = (float*)d_ws;
    float* dis = ws;
    float* inv = ws + NN;
    float* h   = ws + 2 * NN;
    float* t   = h + (size_t)NN * HH;
    float* agg = t + (size_t)NN * HH;

    float* emb_out  = (float*)d_out;                       // N*64
    float* prob_out = (float*)d_out + (size_t)NN * HH;     // N*2

    const int gN    = (NN + 255) / 256;
    const int gE    = (EE + 255) / 256;
    const int gNH4  = (NN * (HH / 4) + 255) / 256;
    const int gSc   = (EE * 16 + 255) / 256;
    const int gGemm = (NN + 31) / 32;

    // degree -> dis/inv
    fill_zero_kernel<<<gN, 256, 0, stream>>>(dis, NN);
    degree_kernel<<<gE, 256, 0, stream>>>(dst, dis, EE);
    norm_kernel<<<gN, 256, 0, stream>>>(dis, inv, NN);

    const float* Ws[3] = {W1, W2, W3};
    const float* bs[3] = {b1, b2, b3};
    const float* hcur = x;
    for (int l = 0; l < 3; ++l) {
        gemm64_wmma_kernel<<<gGemm, 256, 0, stream>>>(hcur, Ws[l], nullptr, t, NN);
        fill_zero4_kernel<<<gNH4, 256, 0, stream>>>((float4*)agg, NN * (HH / 4));
        scatter_kernel<<<gSc, 256, 0, stream>>>(src, dst, dis, t, agg, EE);
        finalize_kernel<<<gNH4, 256, 0, stream>>>((const float4*)t, (const float4*)agg,
                                                  inv, (const float4*)bs[l], (float4*)h,
                                                  (l == 2) ? (float4*)emb_out
                                                           : (float4*)nullptr, NN);
        hcur = h;
    }

    // post-mp: y1 = h@Wp1 + bp1 (reuse t), then 64->2 + log_softmax
    gemm64_wmma_kernel<<<gGemm, 256, 0, stream>>>(h, Wp1, bp1, t, NN);
    head_kernel<<<gN, 256, 0, stream>>>(t, Wp2, bp2, prob_out, NN);
}